// GraphConvLayer_9998683865626
// MI455X (gfx1250) — compile-verified
//
#include <hip/hip_runtime.h>

// 2-layer GCN, feature widths 1 -> 64 -> 1.
// Collapsed math (exact, not an approximation):
//   s[c]   = sum_{e: col=c} x[row]*dis[row]*dis[col]   (incl. self loops)
//   h[i,f] = s[i]*W1[f] + b1[f]            <- dense: done with V_WMMA_F32_16X16X4_F32
//   z[i]   = sum_f h[i,f]*W2[f]            <- GEMV finish in VALU + shfl_xor reduce
//   out[c] = sum_{e: col=c} z[row]*dis[row]*dis[col] + b2
// Bandwidth-bound: 3 passes over the 3.2M-edge list (~64 MB streaming) with
// gathers/atomics into 400 KB node arrays (L2 resident). WMMA handles the only
// dense stage in full f32 precision.

typedef float v2f __attribute__((ext_vector_type(2)));
typedef float v8f __attribute__((ext_vector_type(8)));

// ---------------- node-wise kernels ----------------

__global__ void k_init_deg(float* __restrict__ deg, int n) {
  int i = blockIdx.x * blockDim.x + threadIdx.x;
  if (i < n) deg[i] = 1.0f;  // self-loop contributes 1 to every node's degree
}

__global__ void k_count_deg(const int* __restrict__ col, float* __restrict__ deg, int e) {
  int i = blockIdx.x * blockDim.x + threadIdx.x;
  if (i < e) atomicAdd(&deg[col[i]], 1.0f);
}

// dis = deg^-1/2 ; xd = x*dis (pre-scaled gather operand) ; s seeded with the
// self-loop term x[i]*dis[i]^2 (norm of edge (i,i) is dis[i]*dis[i]).
__global__ void k_dis(const float* __restrict__ x, const float* __restrict__ deg,
                      float* __restrict__ dis, float* __restrict__ xd,
                      float* __restrict__ s, int n) {
  int i = blockIdx.x * blockDim.x + threadIdx.x;
  if (i < n) {
    float d = rsqrtf(deg[i]);  // deg >= 1 always (self loops), matches where(deg>0,...)
    dis[i] = d;
    float v = x[i] * d;
    xd[i] = v;
    s[i] = v * d;
  }
}

__global__ void k_agg1(const int* __restrict__ row, const int* __restrict__ col,
                       const float* __restrict__ xd, const float* __restrict__ dis,
                       float* __restrict__ s, int e) {
  int i = blockIdx.x * blockDim.x + threadIdx.x;
  if (i < e) {
    int r = row[i], c = col[i];
    atomicAdd(&s[c], xd[r] * dis[c]);
  }
}

// ---------------- dense hidden layer via WMMA ----------------
// One wave per 16-node tile. A[16x4]: col0 = s tile, col1 = ones, cols 2,3 = 0.
// B_j[4x16]: row0 = W1[16j..], row1 = b1[16j..], rows 2,3 = 0.
// D_j = A x B_j gives h columns 16j..16j+15 (f32 accumulate on tensor core).
// A layout (ISA 7.12.2, 32-bit A 16x4): lanes 0-15 hold K=0(V0),K=1(V1); lanes
// 16-31 hold K=2(V0),K=3(V1) -> upper lanes carry the zero K-columns.
// B layout mirrors: lanes 0-15 = rows K=0/K=1, lanes 16-31 = rows K=2/K=3.
// D layout: VGPR v, lane L -> h[M = v + 8*(L>>4), N = L&15] (+16j column offset).
//
// NOTE: all lanes issue unconditional loads (indices clamped in-bounds) and the
// zeroing of the unused K-rows/columns is done with value selects (v_cndmask),
// NOT load predication — this avoids s_and_saveexec/exec-restore sequences
// around the WMMAs and keeps the load stream back-to-back.
__global__ void k_hidden_wmma(const float* __restrict__ s,
                              const float* __restrict__ W1,
                              const float* __restrict__ b1,
                              const float* __restrict__ W2,
                              float* __restrict__ z, int n) {
  int lane = threadIdx.x & 31;
  int wave = threadIdx.x >> 5;
  int tile = blockIdx.x * (blockDim.x >> 5) + wave;
  int base = tile * 16;
  if (base >= n) return;  // wave-uniform: EXEC stays all-ones for live waves

  int lo = lane & 15;
  int hi = lane >> 4;

  // Unconditional, clamped load of the s tile; zero via select, not exec.
  int sidx = base + lo;
  bool in = sidx < n;
  if (!in) sidx = n - 1;
  float sv = s[sidx];

  v2f a;
  a.x = (hi == 0 && in) ? sv : 0.0f;  // K=0: s column (upper lanes: K=2 = 0)
  a.y = (hi == 0) ? 1.0f : 0.0f;      // K=1: ones column (upper: K=3 = 0)

  v8f h[4];
  float w2v[4];
#pragma unroll
  for (int j = 0; j < 4; ++j) {
    // j*16+lo <= 63: in-bounds for every lane -> plain coalesced loads.
    float w1v = W1[j * 16 + lo];
    float b1v = b1[j * 16 + lo];
    w2v[j] = W2[j * 16 + lo];
    v2f b;
    b.x = (hi == 0) ? w1v : 0.0f;  // row K=0 = W1 chunk (upper lanes: K=2 = 0)
    b.y = (hi == 0) ? b1v : 0.0f;  // row K=1 = b1 chunk (upper lanes: K=3 = 0)
    v8f c = {};
    h[j] = __builtin_amdgcn_wmma_f32_16x16x4_f32(
        /*neg_a=*/false, a, /*neg_b=*/false, b,
        /*c_mod=*/(short)0, c, /*reuse_a=*/false, /*reuse_b=*/false);
  }

  // z[M] = sum_N h[M,N]*W2[N]; lane holds N = {lo,16+lo,32+lo,48+lo} for 8 Ms.
  float p[8];
#pragma unroll
  for (int v = 0; v < 8; ++v) {
    float acc = 0.0f;
#pragma unroll
    for (int j = 0; j < 4; ++j) acc = fmaf(h[j][v], w2v[j], acc);
#pragma unroll
    for (int m = 1; m < 16; m <<= 1) acc += __shfl_xor(acc, m, 32);  // 16-lane butterfly
    p[v] = acc;
  }

  if (lo < 8) {  // lanes 0-7 write M=0..7, lanes 16-23 write M=8..15
    float val;
    switch (lo) {
      case 0: val = p[0]; break;
      case 1: val = p[1]; break;
      case 2: val = p[2]; break;
      case 3: val = p[3]; break;
      case 4: val = p[4]; break;
      case 5: val = p[5]; break;
      case 6: val = p[6]; break;
      default: val = p[7]; break;
    }
    int m = base + 8 * hi + lo;
    if (m < n) z[m] = val;
  }
}

// out seeded with b2 + self-loop term; zd = z*dis pre-scaled for the gather.
__global__ void k_prep2(const float* __restrict__ z, const float* __restrict__ dis,
                        const float* __restrict__ b2, float* __restrict__ out,
                        float* __restrict__ zd, int n) {
  int i = blockIdx.x * blockDim.x + threadIdx.x;
  if (i < n) {
    float d = dis[i];
    float zi = z[i];
    out[i] = b2[0] + zi * d * d;
    zd[i] = zi * d;
  }
}

__global__ void k_agg2(const int* __restrict__ row, const int* __restrict__ col,
                       const float* __restrict__ zd, const float* __restrict__ dis,
                       float* __restrict__ out, int e) {
  int i = blockIdx.x * blockDim.x + threadIdx.x;
  if (i < e) {
    int r = row[i], c = col[i];
    atomicAdd(&out[c], zd[r] * dis[c]);
  }
}

extern "C" void kernel_launch(void* const* d_in, const int* in_sizes, int n_in,
                              void* d_out, int out_size, void* d_ws, size_t ws_size,
                              hipStream_t stream) {
  const float* x  = (const float*)d_in[0];
  const int*   ei = (const int*)d_in[1];   // edge_index [2, E] int32
  const float* W1 = (const float*)d_in[2]; // [1,64]
  const float* b1 = (const float*)d_in[3]; // [64]
  const float* W2 = (const float*)d_in[4]; // [64,1]
  const float* b2 = (const float*)d_in[5]; // [1]
  float* out = (float*)d_out;

  int n = in_sizes[0];          // 100000 nodes
  int e = in_sizes[1] / 2;      // 3.2M edges
  const int* row = ei;
  const int* col = ei + e;

  // workspace: 6 float arrays of n (deg, dis, xd, s, z, zd) = 2.4 MB
  float* ws  = (float*)d_ws;
  float* deg = ws + 0 * (size_t)n;
  float* dis = ws + 1 * (size_t)n;
  float* xd  = ws + 2 * (size_t)n;
  float* s   = ws + 3 * (size_t)n;
  float* z   = ws + 4 * (size_t)n;
  float* zd  = ws + 5 * (size_t)n;

  const int B = 256;
  int gn = (n + B - 1) / B;
  int ge = (e + B - 1) / B;

  k_init_deg<<<gn, B, 0, stream>>>(deg, n);
  k_count_deg<<<ge, B, 0, stream>>>(col, deg, e);
  k_dis<<<gn, B, 0, stream>>>(x, deg, dis, xd, s, n);
  k_agg1<<<ge, B, 0, stream>>>(row, col, xd, dis, s, e);

  int tiles = (n + 15) / 16;          // one wave (32 lanes) per 16-node tile
  int gw = (tiles + 7) / 8;           // 8 waves per 256-thread block
  k_hidden_wmma<<<gw, B, 0, stream>>>(s, W1, b1, W2, z, n);

  k_prep2<<<gn, B, 0, stream>>>(z, dis, b2, out, zd, n);
  k_agg2<<<ge, B, 0, stream>>>(row, col, zd, dis, out, e);
}